// FIURIModule_32658931319601
// MI455X (gfx1250) — compile-verified
//
#include <hip/hip_runtime.h>
#include <hip/hip_bf16.h>

#define CLAMP_MIN_F (-10.0f)
#define CLAMP_MAX_F (10.0f)

// Native CDNA5 FP32 global atomic add, no-return form, GVS addressing:
// SGPR-pair base + 32-bit VGPR byte offset, device scope so the RMW executes
// at L2 (the 3.2MB accumulator is L2-resident). Guaranteed native instruction.
__device__ __forceinline__ void gatomic_add_f32(float* base, int byte_off, float v) {
    asm volatile("global_atomic_add_f32 %0, %1, %2 scope:SCOPE_DEV"
                 :: "v"(byte_off), "v"(v), "s"(base)
                 : "memory");
}

template <int BB>
__device__ __forceinline__ void process_edge(const float* __restrict__ o_pre,
                                             const float* __restrict__ E,
                                             float* __restrict__ chem,
                                             int s, int d, float wj, int N) {
    float Oj[BB], En[BB];
#pragma unroll
    for (int b = 0; b < BB; ++b) {
        // 32-bit int indexing: lets the compiler use saddr + 32-bit voffset
        Oj[b] = o_pre[b * N + s];
        En[b] = E[b * N + d];
    }
#pragma unroll
    for (int b = 0; b < BB; ++b) {
        float diff = Oj[b] - En[b];
        float sg   = diff > 0.0f ? 1.0f : (diff < 0.0f ? -1.0f : 0.0f);
        float contrib = Oj[b] * wj * sg;
        gatomic_add_f32(chem, (b * N + d) * 4, contrib);
    }
}

template <int BB>
__global__ __launch_bounds__(256) void edge_scatter_kernel(
        const float* __restrict__ o_pre, const float* __restrict__ E,
        const int* __restrict__ src, const int* __restrict__ dst,
        const float* __restrict__ w, float* __restrict__ chem,
        int edges, int N) {
    int t  = blockIdx.x * blockDim.x + threadIdx.x;
    int e0 = t * 4;
    if (e0 >= edges) return;

    // Stay ahead of the streaming index/weight arrays (speculative prefetch:
    // OOB translation failures are silently dropped on gfx1250).
    __builtin_prefetch(src + e0 + 8192, 0, 1);
    __builtin_prefetch(dst + e0 + 8192, 0, 1);
    __builtin_prefetch(w   + e0 + 8192, 0, 1);

    if (e0 + 4 <= edges) {
        const int4   s4 = *reinterpret_cast<const int4*>(src + e0);
        const int4   d4 = *reinterpret_cast<const int4*>(dst + e0);
        const float4 w4 = *reinterpret_cast<const float4*>(w + e0);
        process_edge<BB>(o_pre, E, chem, s4.x, d4.x, w4.x, N);
        process_edge<BB>(o_pre, E, chem, s4.y, d4.y, w4.y, N);
        process_edge<BB>(o_pre, E, chem, s4.z, d4.z, w4.z, N);
        process_edge<BB>(o_pre, E, chem, s4.w, d4.w, w4.w, N);
    } else {
        for (int e = e0; e < edges; ++e) {
            process_edge<BB>(o_pre, E, chem, src[e], dst[e], w[e], N);
        }
    }
}

// Fallback if B != 8 at runtime (reference fixes B=8; keep it general).
__global__ __launch_bounds__(256) void edge_scatter_kernel_dyn(
        const float* __restrict__ o_pre, const float* __restrict__ E,
        const int* __restrict__ src, const int* __restrict__ dst,
        const float* __restrict__ w, float* __restrict__ chem,
        int edges, int N, int B) {
    int e = blockIdx.x * blockDim.x + threadIdx.x;
    if (e >= edges) return;
    int s = src[e], d = dst[e];
    float wj = w[e];
    for (int b = 0; b < B; ++b) {
        float Oj = o_pre[b * N + s];
        float En = E[b * N + d];
        float diff = Oj - En;
        float sg   = diff > 0.0f ? 1.0f : (diff < 0.0f ? -1.0f : 0.0f);
        gatomic_add_f32(chem, (b * N + d) * 4, Oj * wj * sg);
    }
}

__device__ __forceinline__ void node_elem(float e, float c, float t, float dk,
                                          float& no, float& ne) {
    float S = e + c;
    S = fminf(fmaxf(S, CLAMP_MIN_F), CLAMP_MAX_F);
    bool  gt   = S > t;
    float no_  = fmaxf(S - t, 0.0f);
    bool  mask = (!gt) && (S == e);
    no = no_;
    ne = gt ? no_ : (mask ? (e - dk) : S);
}

__global__ __launch_bounds__(256) void node_update_kernel(
        const float* __restrict__ E, const float* __restrict__ chem,
        const float* __restrict__ thr, const float* __restrict__ dec,
        float* __restrict__ new_o, float* __restrict__ new_e,
        int total, int N) {
    int i = (blockIdx.x * blockDim.x + threadIdx.x) * 4;
    if (i >= total) return;
    if ((i + 4 <= total) && ((N & 3) == 0)) {
        int n = i % N;  // N % 4 == 0 -> quad never crosses a batch row
        float4 Ev = *reinterpret_cast<const float4*>(E + i);
        float4 Cv = *reinterpret_cast<const float4*>(chem + i);
        float4 Tv = *reinterpret_cast<const float4*>(thr + n);
        float4 Dv = *reinterpret_cast<const float4*>(dec + n);
        float4 Ov, Nv;
        node_elem(Ev.x, Cv.x, Tv.x, Dv.x, Ov.x, Nv.x);
        node_elem(Ev.y, Cv.y, Tv.y, Dv.y, Ov.y, Nv.y);
        node_elem(Ev.z, Cv.z, Tv.z, Dv.z, Ov.z, Nv.z);
        node_elem(Ev.w, Cv.w, Tv.w, Dv.w, Ov.w, Nv.w);
        *reinterpret_cast<float4*>(new_o + i) = Ov;
        *reinterpret_cast<float4*>(new_e + i) = Nv;
    } else {
        int end = (i + 4 < total) ? (i + 4) : total;
        for (int k = i; k < end; ++k) {
            int n = k % N;
            float no, ne;
            node_elem(E[k], chem[k], thr[n], dec[n], no, ne);
            new_o[k] = no;
            new_e[k] = ne;
        }
    }
}

extern "C" void kernel_launch(void* const* d_in, const int* in_sizes, int n_in,
                              void* d_out, int out_size, void* d_ws, size_t ws_size,
                              hipStream_t stream) {
    const float* chem_in = (const float*)d_in[0];
    const float* E       = (const float*)d_in[1];
    // d_in[2] (O) is unused by the reference computation.
    const int*   src     = (const int*)d_in[3];
    const int*   dst     = (const int*)d_in[4];
    const float* w       = (const float*)d_in[5];
    const float* o_pre   = (const float*)d_in[6];
    const float* thr     = (const float*)d_in[7];
    const float* dec     = (const float*)d_in[8];

    const int N     = in_sizes[7];          // threshold is (N,)
    const int BN    = in_sizes[1];          // E is (B, N)
    const int B     = BN / N;
    const int edges = in_sizes[3];

    float* chem_ws = (float*)d_ws;          // B*N float accumulator
    float* out_o   = (float*)d_out;
    float* out_e   = (float*)d_out + BN;

    // Seed the accumulator with chem_influence (graph-capture-safe async d2d).
    hipMemcpyAsync(chem_ws, chem_in, (size_t)BN * sizeof(float),
                   hipMemcpyDeviceToDevice, stream);

    dim3 blk(256);
    if (B == 8) {
        int ethreads = (edges + 3) / 4;
        dim3 egrd((ethreads + 255) / 256);
        edge_scatter_kernel<8><<<egrd, blk, 0, stream>>>(
            o_pre, E, src, dst, w, chem_ws, edges, N);
    } else {
        dim3 egrd((edges + 255) / 256);
        edge_scatter_kernel_dyn<<<egrd, blk, 0, stream>>>(
            o_pre, E, src, dst, w, chem_ws, edges, N, B);
    }

    int nthreads = (BN + 3) / 4;
    dim3 ngrd((nthreads + 255) / 256);
    node_update_kernel<<<ngrd, blk, 0, stream>>>(
        E, chem_ws, thr, dec, out_o, out_e, BN, N);
}